// GCN_27762668601904
// MI455X (gfx1250) — compile-verified
//
#include <hip/hip_runtime.h>
#include <hip/hip_bf16.h>

#define GCN_N_NODES 50000
#define GCN_N_EDGES 800000
#define GCN_IN      128
#define GCN_H       128
#define GCN_C       32

typedef __attribute__((ext_vector_type(2))) float v2f;
typedef __attribute__((ext_vector_type(8))) float v8f;

// ---------------------------------------------------------------------------
// K1: degree accumulation (float atomics -> global_atomic_add_f32)
// ---------------------------------------------------------------------------
__global__ __launch_bounds__(256) void gcn_deg_kernel(
    const int* __restrict__ src, const int* __restrict__ dst,
    float* __restrict__ deg_out, float* __restrict__ deg_in, int E) {
  int e = blockIdx.x * blockDim.x + threadIdx.x;
  if (e < E) {
    atomicAdd(&deg_out[src[e]], 1.0f);
    atomicAdd(&deg_in[dst[e]], 1.0f);
  }
}

// ---------------------------------------------------------------------------
// K2: deg -> D^{-1/2} in place (zero degree -> 0, matching DGL clamp)
// ---------------------------------------------------------------------------
__global__ __launch_bounds__(256) void gcn_invsqrt_kernel(
    float* __restrict__ deg_out, float* __restrict__ deg_in, int N) {
  int n = blockIdx.x * blockDim.x + threadIdx.x;
  if (n < N) {
    float d = deg_out[n];
    deg_out[n] = (d > 0.0f) ? rsqrtf(d) : 0.0f;
    d = deg_in[n];
    deg_in[n] = (d > 0.0f) ? rsqrtf(d) : 0.0f;
  }
}

// ---------------------------------------------------------------------------
// K3: edge-parallel SpMM scatter.  8 edges per 256-thread block; each group
// of 32 lanes handles one edge, each lane moves a float4 (32*4 = 128 feats).
// Fuses the inv_sqrt_out row scaling into the gather.
// ---------------------------------------------------------------------------
__global__ __launch_bounds__(256) void gcn_spmm_scatter(
    const float* __restrict__ x, const int* __restrict__ src,
    const int* __restrict__ dst, const float* __restrict__ inv_out,
    float* __restrict__ agg, int E) {
  int lane = threadIdx.x & 31;
  int e = blockIdx.x * 8 + (threadIdx.x >> 5);
  if (e >= E) return;
  int s = src[e];
  int d = dst[e];
  float sc = inv_out[s];
  const float4* xr = (const float4*)(x + (size_t)s * 128);
  float4 v = xr[lane];                      // coalesced 512B per edge-group
  float* ar = agg + (size_t)d * 128 + lane * 4;
  atomicAdd(ar + 0, v.x * sc);
  atomicAdd(ar + 1, v.y * sc);
  atomicAdd(ar + 2, v.z * sc);
  atomicAdd(ar + 3, v.w * sc);
}

// ---------------------------------------------------------------------------
// K4: GEMM1  h1 = relu( (inv_in .* agg) @ W[128x128] + b )  via f32 WMMA.
// Block = 256 threads = 8 waves; block owns 16 rows, wave w owns cols
// [16w, 16w+16).  A tile staged (pre-scaled) in LDS.
// f32 WMMA fragment layout (ISA 7.12.2):
//   A 16x4 : lanes 0-15 -> M=lane, VGPR0/1 = K=k,k+1 ; lanes 16-31 -> K=k+2,k+3
//   B 4x16 : lanes 0-15 -> N=lane, VGPR0/1 = K=k,k+1 ; lanes 16-31 -> K=k+2,k+3
//   C/D    : VGPR r, lanes 0-15 -> (M=r, N=lane); lanes 16-31 -> (M=r+8)
// ---------------------------------------------------------------------------
__global__ __launch_bounds__(256) void gcn_gemm1_wmma(
    const float* __restrict__ agg, const float* __restrict__ inv_in,
    const float* __restrict__ W, const float* __restrict__ bias,
    float* __restrict__ out) {
  __shared__ float As[16][132];             // +4 pad: kill LDS bank conflicts
  const int row0 = blockIdx.x * 16;         // 3125 * 16 == 50000 exactly
  const int tid  = threadIdx.x;

  for (int i = tid; i < 16 * 128; i += 256) {
    int r = i >> 7, c = i & 127;
    As[r][c] = agg[(size_t)(row0 + r) * 128 + c] * inv_in[row0 + r];
  }
  __syncthreads();

  const int wave = tid >> 5;
  const int lane = tid & 31;
  const int col0 = wave * 16;
  const int half = lane >> 4;               // 0: K=k,k+1   1: K=k+2,k+3
  const int mrow = lane & 15;               // M row (A) / N col (B)

  v8f acc = {};
  for (int kk = 0; kk < 128; kk += 4) {
    int k2 = kk + half * 2;
    v2f a, b;
    a.x = As[mrow][k2];
    a.y = As[mrow][k2 + 1];
    b.x = W[(size_t)k2 * 128 + col0 + mrow];
    b.y = W[(size_t)(k2 + 1) * 128 + col0 + mrow];
    acc = __builtin_amdgcn_wmma_f32_16x16x4_f32(
        /*neg_a=*/false, a, /*neg_b=*/false, b,
        /*c_mod=*/(short)0, acc, /*reuse_a=*/false, /*reuse_b=*/false);
  }

  float bv = bias[col0 + mrow];
  for (int r = 0; r < 8; ++r) {
    int row = row0 + r + half * 8;
    float v = acc[r] + bv;
    out[(size_t)row * 128 + col0 + mrow] = (v > 0.0f) ? v : 0.0f;
  }
}

// ---------------------------------------------------------------------------
// K5: GEMM2  out = (inv_in .* agg) @ W[128x32] + b   (no relu).
// Block = 256 threads = 8 waves; block owns 64 rows; wave -> (rowgroup, coltile).
// ---------------------------------------------------------------------------
__global__ __launch_bounds__(256) void gcn_gemm2_wmma(
    const float* __restrict__ agg, const float* __restrict__ inv_in,
    const float* __restrict__ W, const float* __restrict__ bias,
    float* __restrict__ out, int N) {
  __shared__ float As[64][132];             // ~33 KB of the 320 KB WGP LDS
  const int row0 = blockIdx.x * 64;
  const int tid  = threadIdx.x;

  for (int i = tid; i < 64 * 128; i += 256) {
    int r = i >> 7, c = i & 127;
    int row = row0 + r;
    As[r][c] = (row < N) ? agg[(size_t)row * 128 + c] * inv_in[row] : 0.0f;
  }
  __syncthreads();

  const int wave = tid >> 5;
  const int lane = tid & 31;
  const int rg   = wave >> 1;               // 0..3 : 16-row group
  const int col0 = (wave & 1) * 16;         // 0 or 16
  const int half = lane >> 4;
  const int mrow = lane & 15;

  v8f acc = {};
  for (int kk = 0; kk < 128; kk += 4) {
    int k2 = kk + half * 2;
    v2f a, b;
    a.x = As[rg * 16 + mrow][k2];
    a.y = As[rg * 16 + mrow][k2 + 1];
    b.x = W[(size_t)k2 * 32 + col0 + mrow];
    b.y = W[(size_t)(k2 + 1) * 32 + col0 + mrow];
    acc = __builtin_amdgcn_wmma_f32_16x16x4_f32(
        false, a, false, b, (short)0, acc, false, false);
  }

  float bv = bias[col0 + mrow];
  for (int r = 0; r < 8; ++r) {
    int row = row0 + rg * 16 + r + half * 8;
    if (row < N) out[(size_t)row * 32 + col0 + mrow] = acc[r] + bv;
  }
}

// ---------------------------------------------------------------------------
// Launch: deg -> invsqrt -> [zero agg, scatter, gemm] x 2
// ---------------------------------------------------------------------------
extern "C" void kernel_launch(void* const* d_in, const int* in_sizes, int n_in,
                              void* d_out, int out_size, void* d_ws, size_t ws_size,
                              hipStream_t stream) {
  const float* in_feat = (const float*)d_in[0];
  const int*   src     = (const int*)d_in[1];
  const int*   dst     = (const int*)d_in[2];
  const float* W1      = (const float*)d_in[3];
  const float* b1      = (const float*)d_in[4];
  const float* W2      = (const float*)d_in[5];
  const float* b2      = (const float*)d_in[6];
  float* out = (float*)d_out;

  const int N = GCN_N_NODES;
  const int E = GCN_N_EDGES;

  // Workspace layout (floats): [inv_out N][inv_in N][agg N*128][h1 N*128]
  float* inv_out = (float*)d_ws;
  float* inv_in  = inv_out + N;
  float* agg     = inv_in + N;
  float* h1      = agg + (size_t)N * 128;

  // degrees
  hipMemsetAsync(inv_out, 0, (size_t)2 * N * sizeof(float), stream);
  gcn_deg_kernel<<<(E + 255) / 256, 256, 0, stream>>>(src, dst, inv_out, inv_in, E);
  gcn_invsqrt_kernel<<<(N + 255) / 256, 256, 0, stream>>>(inv_out, inv_in, N);

  // layer 1
  hipMemsetAsync(agg, 0, (size_t)N * 128 * sizeof(float), stream);
  gcn_spmm_scatter<<<(E + 7) / 8, 256, 0, stream>>>(in_feat, src, dst, inv_out, agg, E);
  gcn_gemm1_wmma<<<N / 16, 256, 0, stream>>>(agg, inv_in, W1, b1, h1);

  // layer 2 (reuse agg buffer)
  hipMemsetAsync(agg, 0, (size_t)N * 128 * sizeof(float), stream);
  gcn_spmm_scatter<<<(E + 7) / 8, 256, 0, stream>>>(h1, src, dst, inv_out, agg, E);
  gcn_gemm2_wmma<<<(N + 63) / 64, 256, 0, stream>>>(agg, inv_in, W2, b2, out, N);
}